// EfficientVMamba_MambaBlock_86569360818677
// MI455X (gfx1250) — compile-verified
//
#include <hip/hip_runtime.h>
#include <hip/hip_bf16.h>

typedef __attribute__((ext_vector_type(2))) float v2f;
typedef __attribute__((ext_vector_type(8))) float v8f;
typedef int v4i_ __attribute__((vector_size(16)));   // matches builtin's pointee type

#define TILE_M 128
#define TILE_N 128
#define TILE_K 16
#define LDT 20   // LDS row pitch (floats): 80B rows -> float4-aligned, conflict-free frag reads

// -------- CDNA5 async global->LDS copy (ASYNCcnt-tracked DMA) ---------------
#if defined(__has_builtin)
#  if __has_builtin(__builtin_amdgcn_global_load_async_to_lds_b128)
#    define HAVE_ASYNC 1
#  endif
#endif

#ifdef HAVE_ASYNC
#  define ASYNC_LOAD_B128(gp, lp)                                              \
      __builtin_amdgcn_global_load_async_to_lds_b128(                          \
          (__attribute__((address_space(1))) v4i_*)(gp),                       \
          (__attribute__((address_space(3))) v4i_*)(lp), 0, 0)
#  if __has_builtin(__builtin_amdgcn_s_wait_asynccnt)
#    define WAIT_ASYNC() __builtin_amdgcn_s_wait_asynccnt(0)
#  else
#    define WAIT_ASYNC() asm volatile("s_wait_asynccnt 0x0" ::: "memory")
#  endif
#else
#  define WAIT_ASYNC()
#endif

__device__ __forceinline__ float sigmoidf_(float v) { return 1.f / (1.f + __expf(-v)); }
__device__ __forceinline__ float siluf_(float v)    { return v * sigmoidf_(v); }
__device__ __forceinline__ float softplusf_(float v){ return v > 20.f ? v : log1pf(__expf(v)); }

// ---------------------------------------------------------------------------
// Generic fp32 WMMA GEMM: C[m][n] = sum_k A[m][k] * W[n][k]  (+ epilogue)
// MODE 0: C = acc + bias[n]                      (plain row-major store)
// MODE 1: cross_scan permuted store (+bias)      (row m=b*1024+s -> (b,k,l))
// MODE 2: C = softplus(acc + bias_k[n])          (per-direction W/bias)
// MODE 3: C = xres + gate_b[n]*(acc + bias[n] + skip)
// wstride != 0 => W (and bias via bstride) indexed by direction k=(m0/256)&3
// Double-buffered LDS; tiles staged with GLOBAL_LOAD_ASYNC_TO_LDS_B128 when
// available (one s_wait_asynccnt + one barrier per K tile).
// ---------------------------------------------------------------------------
template<int MODE>
__global__ __launch_bounds__(256)
void gemm_wmma_kernel(const float* __restrict__ A, int lda,
                      const float* __restrict__ W, int ldw, long long wstride,
                      const float* __restrict__ bias, long long bstride,
                      float* __restrict__ C, int ldc,
                      int M, int N, int K,
                      const float* __restrict__ skip,
                      const float* __restrict__ gate,
                      const float* __restrict__ xres)
{
    __shared__ __align__(16) float As[2][TILE_M * LDT];
    __shared__ __align__(16) float Bs[2][TILE_N * LDT];

    const int tid = threadIdx.x;
    const int m0 = blockIdx.y * TILE_M;
    const int n0 = blockIdx.x * TILE_N;

    int kdir = 0;
    if (wstride != 0) { kdir = (m0 >> 8) & 3; W += (long long)kdir * wstride; }
    const float* biasp = bias ? (bias + (long long)kdir * bstride) : nullptr;

    const int wave = tid >> 5;
    const int lane = tid & 31;
    const int wm = (wave >> 2) * 64;   // wave M offset: 0 / 64
    const int wn = (wave & 3) * 32;    // wave N offset: 0,32,64,96
    const int lm = lane & 15;          // fragment row/col within 16
    const int kb = (lane >> 4) * 2;    // K pair: 0 or 2

    v8f zero = {};
    v8f acc[4][2];
#pragma unroll
    for (int i = 0; i < 4; ++i)
#pragma unroll
        for (int j = 0; j < 2; ++j) acc[i][j] = zero;

    // stage one 128x16 A tile + 128x16 B tile into LDS buffer `buf`
    auto stage = [&](int k0, int buf) {
#pragma unroll
        for (int p = 0; p < 2; ++p) {
            int idx = p * 1024 + tid * 4;
            int r  = idx >> 4;
            int cc = idx & 15;
            int gm = m0 + r;
            int gn = n0 + r;
#ifdef HAVE_ASYNC
            if (gm < M)
                ASYNC_LOAD_B128(A + (long long)gm * lda + (k0 + cc),
                                &As[buf][r * LDT + cc]);
            else
                *(float4*)(&As[buf][r * LDT + cc]) = make_float4(0.f, 0.f, 0.f, 0.f);
            if (gn < N)
                ASYNC_LOAD_B128(W + (long long)gn * ldw + (k0 + cc),
                                &Bs[buf][r * LDT + cc]);
            else
                *(float4*)(&Bs[buf][r * LDT + cc]) = make_float4(0.f, 0.f, 0.f, 0.f);
#else
            float4 av = make_float4(0.f, 0.f, 0.f, 0.f);
            if (gm < M) av = *(const float4*)(A + (long long)gm * lda + (k0 + cc));
            *(float4*)(&As[buf][r * LDT + cc]) = av;
            float4 bv = make_float4(0.f, 0.f, 0.f, 0.f);
            if (gn < N) bv = *(const float4*)(W + (long long)gn * ldw + (k0 + cc));
            *(float4*)(&Bs[buf][r * LDT + cc]) = bv;
#endif
        }
    };

    const int nk = K / TILE_K;
    stage(0, 0);
    for (int t = 0; t < nk; ++t) {
        WAIT_ASYNC();            // own async tile copies complete (no-op in fallback)
        __syncthreads();         // all waves' tile data visible; prior reads done
        if (t + 1 < nk) stage((t + 1) * TILE_K, (t + 1) & 1);
        const int buf = t & 1;
#pragma unroll
        for (int kk = 0; kk < TILE_K; kk += 4) {
            v2f af[4], bfr[2];
#pragma unroll
            for (int i = 0; i < 4; ++i)
                af[i] = *(const v2f*)(&As[buf][(wm + i * 16 + lm) * LDT + kk + kb]);
#pragma unroll
            for (int j = 0; j < 2; ++j)
                bfr[j] = *(const v2f*)(&Bs[buf][(wn + j * 16 + lm) * LDT + kk + kb]);
#pragma unroll
            for (int i = 0; i < 4; ++i)
#pragma unroll
                for (int j = 0; j < 2; ++j)
                    acc[i][j] = __builtin_amdgcn_wmma_f32_16x16x4_f32(
                        false, af[i], false, bfr[j], (short)0, acc[i][j], false, false);
        }
    }

    // Epilogue. C/D frag layout: M = v + 8*(lane>=16), N = lane&15.
    const int mo = (lane >> 4) * 8;
    const int nloc = lane & 15;
#pragma unroll
    for (int i = 0; i < 4; ++i) {
#pragma unroll
        for (int j = 0; j < 2; ++j) {
            int gn = n0 + wn + j * 16 + nloc;
            if (gn >= N) continue;
            float bval = biasp ? biasp[gn] : 0.f;
#pragma unroll
            for (int v = 0; v < 8; ++v) {
                int gm = m0 + wm + i * 16 + v + mo;
                if (gm >= M) continue;
                float r = acc[i][j][v] + bval;
                if (MODE == 0) {
                    C[(long long)gm * ldc + gn] = r;
                } else if (MODE == 1) {
                    // cross_scan permuted store: gm = b*1024 + s, s -> (k,l)
                    int b = gm >> 10, s = gm & 1023;
                    int gi = s >> 5, gj = s & 31;
                    int kq, l;
                    if (!(gi & 1) && !(gj & 1))      { kq = 0; l = (gi >> 1) * 16 + (gj >> 1); }
                    else if ((gi & 1) && !(gj & 1))  { kq = 1; l = (gj >> 1) * 16 + (gi >> 1); }
                    else if (!(gi & 1) && (gj & 1))  { kq = 2; l = (gi >> 1) * 16 + (gj >> 1); }
                    else                             { kq = 3; l = (gj >> 1) * 16 + (gi >> 1); }
                    int row = ((b << 2) + kq) * 256 + l;
                    C[(long long)row * ldc + gn] = r;
                } else if (MODE == 2) {
                    C[(long long)gm * ldc + gn] = softplusf_(r);
                } else { // MODE 3: out = x + gate*(acc + bias + skip)
                    int b = gm >> 10;
                    float g = gate[(long long)b * 2304 + gn];
                    C[(long long)gm * ldc + gn] =
                        xres[(long long)gm * ldc + gn] +
                        g * (r + skip[(long long)gm * ldc + gn]);
                }
            }
        }
    }
}

// ---------------------------------------------------------------------------
// silu(c) elementwise
// ---------------------------------------------------------------------------
__global__ __launch_bounds__(256)
void silu_kernel(const float* __restrict__ in, float* __restrict__ outp, int n)
{
    int i = blockIdx.x * 256 + threadIdx.x;
    if (i < n) outp[i] = siluf_(in[i]);
}

// ---------------------------------------------------------------------------
// mod = silu(c) @ W_mod^T + b_mod : tiny (8 x 2304, K=1536) -> plain VALU dot
// ---------------------------------------------------------------------------
__global__ __launch_bounds__(256)
void mod_gemm_kernel(const float* __restrict__ sc, const float* __restrict__ Wm,
                     const float* __restrict__ bm, float* __restrict__ mod)
{
    int o = blockIdx.x * 256 + threadIdx.x;
    if (o >= 8 * 2304) return;
    int b = o / 2304, col = o - b * 2304;
    const float* a  = sc + b * 1536;
    const float* wr = Wm + (long long)col * 1536;
    float s = 0.f;
    for (int j = 0; j < 1536; j += 4) {
        float4 av = *(const float4*)(a + j);
        float4 wv = *(const float4*)(wr + j);
        s = fmaf(av.x, wv.x, s); s = fmaf(av.y, wv.y, s);
        s = fmaf(av.z, wv.z, s); s = fmaf(av.w, wv.w, s);
    }
    mod[o] = s + bm[col];
}

// ---------------------------------------------------------------------------
// Fused: LN1 -> *(1+scale)+shift -> store skip -> LN2 -> store xn
// one block per (b,s) row of 768; 3 elems/thread
// ---------------------------------------------------------------------------
__global__ __launch_bounds__(256)
void ln_mod_kernel(const float* __restrict__ x, const float* __restrict__ mod,
                   const float* __restrict__ g1, const float* __restrict__ b1,
                   const float* __restrict__ g2, const float* __restrict__ b2,
                   float* __restrict__ skip, float* __restrict__ xn)
{
    __shared__ float s1[256], s2[256];
    const int row = blockIdx.x;
    const int b = row >> 10;
    const int tid = threadIdx.x;
    const float* xr = x + (long long)row * 768;
    const float* modb = mod + (long long)b * 2304;

    float v[3]; float sum = 0.f, sq = 0.f;
#pragma unroll
    for (int r = 0; r < 3; ++r) {
        int d = tid + 256 * r;
        v[r] = xr[d]; sum += v[r]; sq = fmaf(v[r], v[r], sq);
    }
    s1[tid] = sum; s2[tid] = sq; __syncthreads();
    for (int off = 128; off > 0; off >>= 1) {
        if (tid < off) { s1[tid] += s1[tid + off]; s2[tid] += s2[tid + off]; }
        __syncthreads();
    }
    float mean = s1[0] * (1.f / 768.f);
    float var  = s2[0] * (1.f / 768.f) - mean * mean;
    float rs = rsqrtf(var + 1e-5f);
    __syncthreads();

    float sum2 = 0.f, sq2 = 0.f;
#pragma unroll
    for (int r = 0; r < 3; ++r) {
        int d = tid + 256 * r;
        float t = (v[r] - mean) * rs * g1[d] + b1[d];
        t = t * (1.f + modb[768 + d]) + modb[d];
        v[r] = t;
        skip[(long long)row * 768 + d] = t;
        sum2 += t; sq2 = fmaf(t, t, sq2);
    }
    s1[tid] = sum2; s2[tid] = sq2; __syncthreads();
    for (int off = 128; off > 0; off >>= 1) {
        if (tid < off) { s1[tid] += s1[tid + off]; s2[tid] += s2[tid + off]; }
        __syncthreads();
    }
    float mean2 = s1[0] * (1.f / 768.f);
    float var2  = s2[0] * (1.f / 768.f) - mean2 * mean2;
    float rs2 = rsqrtf(var2 + 1e-5f);
#pragma unroll
    for (int r = 0; r < 3; ++r) {
        int d = tid + 256 * r;
        xn[(long long)row * 768 + d] = (v[r] - mean2) * rs2 * g2[d] + b2[d];
    }
}

// ---------------------------------------------------------------------------
// Selective scan (4 directions batched), fused with +xs*Dp, cross_merge and
// silu(mamba_x)*y. One thread per (b,k,d); 16 states in registers; L=256 steps.
// Grid: B*4*(E/256) = 192 blocks of 256.
// ---------------------------------------------------------------------------
__global__ __launch_bounds__(256)
void scan_kernel(const float* __restrict__ zs, const float* __restrict__ dlt,
                 const float* __restrict__ dbc, const float* __restrict__ A_log,
                 const float* __restrict__ Dp, const float* __restrict__ mx,
                 float* __restrict__ xo)
{
    const int bid = blockIdx.x;
    const int chunk = bid % 6;
    const int bk = bid / 6;
    const int b = bk >> 2;
    const int k = bk & 3;
    const int d = chunk * 256 + threadIdx.x;   // 0..1535

    float Aa[16];
    const float* al = A_log + ((long long)(k * 1536 + d)) * 16;
#pragma unroll
    for (int n = 0; n < 16; ++n) Aa[n] = -__expf(al[n]);
    const float dp = Dp[k * 1536 + d];

    float h[16];
#pragma unroll
    for (int n = 0; n < 16; ++n) h[n] = 0.f;

    const long long rowbase = (long long)(b * 4 + k) * 256;
    for (int l = 0; l < 256; ++l) {
        const long long row = rowbase + l;
        float dv = dlt[row * 1536 + d];
        float u  = zs[row * 1536 + d];
        const float* bc = dbc + row * 80;   // [0:48]=dr, [48:64]=B, [64:80]=C
        float dtu = dv * u;
        float y = 0.f;
#pragma unroll
        for (int n = 0; n < 16; ++n) {
            float dA = __expf(dv * Aa[n]);
            h[n] = fmaf(dA, h[n], dtu * bc[48 + n]);
            y = fmaf(h[n], bc[64 + n], y);
        }
        // cross_merge: (k,l) -> grid (gi,gj) -> s
        int p = l >> 4, q = l & 15;
        int gi, gj;
        switch (k) {
            case 0:  gi = 2 * p;     gj = 2 * q;     break;
            case 1:  gi = 2 * q + 1; gj = 2 * p;     break;
            case 2:  gi = 2 * p;     gj = 2 * q + 1; break;
            default: gi = 2 * q + 1; gj = 2 * p + 1; break;
        }
        int s = gi * 32 + gj;
        long long oidx = (long long)(b * 1024 + s) * 1536 + d;
        float mv = mx[oidx];
        xo[oidx] = siluf_(mv) * (y + u * dp);
    }
}

// ---------------------------------------------------------------------------
extern "C" void kernel_launch(void* const* d_in, const int* in_sizes, int n_in,
                              void* d_out, int out_size, void* d_ws, size_t ws_size,
                              hipStream_t stream)
{
    (void)in_sizes; (void)n_in; (void)out_size; (void)ws_size;
    const float* x     = (const float*)d_in[0];
    const float* c     = (const float*)d_in[1];
    /* d_in[2] = "w" : unused by the reference body */
    const float* W_mod = (const float*)d_in[3];
    const float* b_mod = (const float*)d_in[4];
    const float* g1    = (const float*)d_in[5];
    const float* b1    = (const float*)d_in[6];
    const float* g2    = (const float*)d_in[7];
    const float* b2    = (const float*)d_in[8];
    const float* Wz    = (const float*)d_in[9];
    const float* bz    = (const float*)d_in[10];
    const float* Wx    = (const float*)d_in[11];
    const float* bx    = (const float*)d_in[12];
    const float* Wconv = (const float*)d_in[13];
    const float* bconv = (const float*)d_in[14];
    const float* W_dbc = (const float*)d_in[15];
    const float* W_dt  = (const float*)d_in[16];
    const float* b_dt  = (const float*)d_in[17];
    const float* A_log = (const float*)d_in[18];
    const float* Dp    = (const float*)d_in[19];
    const float* Wf    = (const float*)d_in[20];
    const float* bfv   = (const float*)d_in[21];

    float* ws   = (float*)d_ws;
    float* sc   = ws;               // silu(c)            : 12288
    float* mod  = ws + 12288;       // (8,2304)           : 18432
    float* skip = ws + 30720;       // (8,1024,768)       : 6291456
    float* xn   = ws + 6322176;     // (8,1024,768)       : 6291456
    float* mx   = ws + 12613632;    // mamba_x (8192,1536): 12582912
    float* z0   = ws + 25196544;    // pre-conv z / xo    : 12582912
    float* zs   = ws + 37779456;    // cross-scanned z    : 12582912
    float* dbc  = ws + 50362368;    // (8192,80)          : 655360
    float* dlt  = ws + 51017728;    // delta (8192,1536)  : 12582912
    float* xo   = z0;               // reuse after zs is produced
    float* out  = (float*)d_out;

    dim3 blk(256);

    silu_kernel<<<48, blk, 0, stream>>>(c, sc, 8 * 1536);
    mod_gemm_kernel<<<72, blk, 0, stream>>>(sc, W_mod, b_mod, mod);
    ln_mod_kernel<<<8192, blk, 0, stream>>>(x, mod, g1, b1, g2, b2, skip, xn);

    // mamba_z = xn @ Wz^T + bz ; mamba_x = xn @ Wx^T + bx
    gemm_wmma_kernel<0><<<dim3(12, 64), blk, 0, stream>>>(
        xn, 768, Wz, 768, 0, bz, 0, z0, 1536, 8192, 1536, 768, nullptr, nullptr, nullptr);
    gemm_wmma_kernel<0><<<dim3(12, 64), blk, 0, stream>>>(
        xn, 768, Wx, 768, 0, bx, 0, mx, 1536, 8192, 1536, 768, nullptr, nullptr, nullptr);

    // mamba_z = mamba_z @ Wconv^T + bconv, stored directly cross-scanned (b,4,L,E)
    gemm_wmma_kernel<1><<<dim3(12, 64), blk, 0, stream>>>(
        z0, 1536, Wconv, 1536, 0, bconv, 0, zs, 1536, 8192, 1536, 1536, nullptr, nullptr, nullptr);

    // dBC = zs @ W_dbc[k]^T    (N=80, per-direction weights)
    gemm_wmma_kernel<0><<<dim3(1, 64), blk, 0, stream>>>(
        zs, 1536, W_dbc, 1536, (long long)80 * 1536, nullptr, 0,
        dbc, 80, 8192, 80, 1536, nullptr, nullptr, nullptr);

    // delta = softplus(dr @ W_dt[k]^T + b_dt[k])   (K=48)
    gemm_wmma_kernel<2><<<dim3(12, 64), blk, 0, stream>>>(
        dbc, 80, W_dt, 48, (long long)1536 * 48, b_dt, 1536,
        dlt, 1536, 8192, 1536, 48, nullptr, nullptr, nullptr);

    // selective scan + cross_merge + silu(mamba_x)*y  -> xo (merged layout)
    scan_kernel<<<192, blk, 0, stream>>>(zs, dlt, dbc, A_log, Dp, mx, xo);

    // out = x + gate * (xo @ Wf^T + bf + skip)
    gemm_wmma_kernel<3><<<dim3(6, 64), blk, 0, stream>>>(
        xo, 1536, Wf, 1536, 0, bfv, 0, out, 768, 8192, 768, 1536,
        skip, mod + 1536, x);
}